// DynamicMaskWord2vec_90640989814977
// MI455X (gfx1250) — compile-verified
//
#include <hip/hip_runtime.h>
#include <hip/hip_bf16.h>
#include <cmath>

typedef float v2f __attribute__((ext_vector_type(2)));
typedef float v8f __attribute__((ext_vector_type(8)));

#define DIM 256
#define BATCH 2048
#define MAX_UNMASKED 64
#define MAX_MASKED 16
#define NUM_NEG 16
#define SLOTS (MAX_MASKED + MAX_MASKED * NUM_NEG)  // 272
#define GROUPS (SLOTS / 16)                        // 17
#define WAVES_PER_BLOCK 8

// ---------------------------------------------------------------------------
// Kernel 1: masked average of unmasked token embeddings -> inp_avg (B, 256)
// One block per batch element, one thread per dim; every iteration reads one
// fully-coalesced 1KB embedding row.
// ---------------------------------------------------------------------------
__global__ __launch_bounds__(DIM) void w2v_avg_kernel(
    const float* __restrict__ target_emb,
    const int* __restrict__ unmasked_ids,
    const int* __restrict__ unmasked_mask,
    float* __restrict__ avg_out) {
  const int b = blockIdx.x;
  const int d = threadIdx.x;  // 0..255
  const int* ids = unmasked_ids + b * MAX_UNMASKED;
  const int* msk = unmasked_mask + b * MAX_UNMASKED;
  float sum = 0.f;
  int cnt = 0;
  for (int u = 0; u < MAX_UNMASKED; ++u) {
    const int m = msk[u];
    const int id = ids[u];
    if (m) {
      sum += target_emb[(size_t)id * DIM + d];
      cnt += 1;
    }
  }
  const float c = (float)(cnt < 1 ? 1 : cnt);
  avg_out[(size_t)b * DIM + d] = sum / c;
}

// log_sigmoid(x) = min(x,0) - log1p(exp(-|x|))
__device__ __forceinline__ float log_sigmoid(float x) {
  return fminf(x, 0.f) - log1pf(expf(-fabsf(x)));
}

// ---------------------------------------------------------------------------
// Kernel 2: scores + loss via V_WMMA_F32_16X16X4_F32.
// One wave (wave32) per batch element. For each group of 16 context rows:
//   A (16x4 f32): lane m<16 loads row_m[4k..4k+1], lane m+16 loads row_m[4k+2..3]
//   B (4x16 f32): all 16 columns = inp_avg chunk, mirrored lane layout from LDS
// After 64 accumulating WMMAs, C[m,n] = dot(row_m, inp_avg) for every n:
// lane 0 holds scores 0..7 in c[0..7], lane 16 holds scores 8..15.
// ---------------------------------------------------------------------------
__global__ __launch_bounds__(32 * WAVES_PER_BLOCK) void w2v_score_kernel(
    const float* __restrict__ context_emb,
    const int* __restrict__ masked_ids,
    const int* __restrict__ neg_ids,
    const float* __restrict__ inp_avg,
    float* __restrict__ partials) {
  __shared__ __align__(16) float s_avg[WAVES_PER_BLOCK * DIM];
  __shared__ float s_block;

  const int tid = threadIdx.x;
  const int wave = tid >> 5;
  const int lane = tid & 31;
  const int lane15 = lane & 15;
  const int hi = lane >> 4;  // 0: K0/K1 half, 1: K2/K3 half
  const int b = blockIdx.x * WAVES_PER_BLOCK + wave;

  // Stage this wave's inp_avg row (1KB) into its LDS slice.
  float* my_avg = s_avg + wave * DIM;
  const float* src = inp_avg + (size_t)b * DIM;
  for (int i = lane; i < DIM; i += 32) my_avg[i] = src[i];
  if (tid == 0) s_block = 0.f;
  __syncthreads();

  const float* bptr = my_avg + 2 * hi;
  float wave_acc = 0.f;

  for (int g = 0; g < GROUPS; ++g) {
    // Slot j for this lane pair: group 0 = the 16 positive (masked) rows,
    // groups 1..16 = the 16 negatives of masked position g-1.
    const int j = g * 16 + lane15;
    const int jm = (j < MAX_MASKED) ? j : (MAX_MASKED - 1);
    int jn = j - MAX_MASKED;
    jn = (jn < 0) ? 0 : jn;
    const int idx_pos = masked_ids[b * MAX_MASKED + jm];
    const int idx_neg = neg_ids[b * (MAX_MASKED * NUM_NEG) + jn];
    const int idx = (j < MAX_MASKED) ? idx_pos : idx_neg;

    const float* aptr = context_emb + (size_t)idx * DIM + 2 * hi;

    v8f c = {};
#pragma unroll 4
    for (int kc = 0; kc < DIM / 4; ++kc) {
      const v2f a = *(const v2f*)(aptr + kc * 4);
      const v2f bb = *(const v2f*)(bptr + kc * 4);
      // (neg_a, A, neg_b, B, c_mod, C, reuse_a, reuse_b)
      c = __builtin_amdgcn_wmma_f32_16x16x4_f32(
          false, a, false, bb, (short)0, c, false, false);
    }

    // pos term: log_sigmoid(s); neg term: log_sigmoid(-s)
    const float sign = (g == 0) ? 1.f : -1.f;
    float s8 = 0.f;
#pragma unroll
    for (int v = 0; v < 8; ++v) s8 += log_sigmoid(sign * c[v]);
    // lanes 0..15 all replicate scores 0..7; lanes 16..31 replicate 8..15:
    // count each slot exactly once via lane 0 and lane 16 (branchless, keeps
    // EXEC all-ones around the WMMA loop).
    wave_acc += (lane15 == 0) ? s8 : 0.f;
  }

  // wave32 reduction
#pragma unroll
  for (int off = 16; off > 0; off >>= 1)
    wave_acc += __shfl_xor(wave_acc, off, 32);
  if (lane == 0) atomicAdd(&s_block, wave_acc);
  __syncthreads();
  if (tid == 0) partials[blockIdx.x] = s_block;
}

// ---------------------------------------------------------------------------
// Kernel 3: fold 256 block partials -> scalar mean loss.
// ---------------------------------------------------------------------------
__global__ __launch_bounds__(256) void w2v_reduce_kernel(
    const float* __restrict__ partials, float* __restrict__ out) {
  __shared__ float s[256];
  const int tid = threadIdx.x;
  s[tid] = partials[tid];
  __syncthreads();
  for (int off = 128; off > 0; off >>= 1) {
    if (tid < off) s[tid] += s[tid + off];
    __syncthreads();
  }
  if (tid == 0) out[0] = -s[0] / (float)(BATCH * MAX_MASKED);
}

extern "C" void kernel_launch(void* const* d_in, const int* in_sizes, int n_in,
                              void* d_out, int out_size, void* d_ws, size_t ws_size,
                              hipStream_t stream) {
  const float* target_emb   = (const float*)d_in[0];
  const float* context_emb  = (const float*)d_in[1];
  const int*   unmasked_ids = (const int*)d_in[2];
  const int*   unmasked_mask= (const int*)d_in[3];
  const int*   masked_ids   = (const int*)d_in[4];
  const int*   neg_ids      = (const int*)d_in[5];

  float* avg      = (float*)d_ws;                 // BATCH*DIM floats (2 MB)
  float* partials = avg + (size_t)BATCH * DIM;    // BATCH/WAVES_PER_BLOCK floats

  w2v_avg_kernel<<<BATCH, DIM, 0, stream>>>(
      target_emb, unmasked_ids, unmasked_mask, avg);

  w2v_score_kernel<<<BATCH / WAVES_PER_BLOCK, 32 * WAVES_PER_BLOCK, 0, stream>>>(
      context_emb, masked_ids, neg_ids, avg, partials);

  w2v_reduce_kernel<<<1, 256, 0, stream>>>(partials, (float*)d_out);
}